// _fasterRCNN_inc_18116172055068
// MI455X (gfx1250) — compile-verified
//
#include <hip/hip_runtime.h>

// Faster-RCNN detection post-processing for MI455X (gfx1250, wave32).
// One workgroup per class (20 blocks). All per-class state in LDS (~59KB of
// the 320KB/WGP). Serial NMS is the bottleneck -> minimize barriers.
// WMMA f32 16x16x4 computes the pairwise (areaBox + areaGT) matrix used in
// the GT-dedup IoU denominator as a rank-2 outer product (exact, x1.0 mults).

#define N_ROIS  2048
#define N_CLS   21
#define N_DET   (N_CLS - 1)
#define MAX_GT  64
#define THREADS 256
#define PER_T   (N_ROIS / THREADS)
#define SCORE_T 0.3f
#define NMS_T   0.3f
#define DEDUP_T 0.3f

typedef __attribute__((ext_vector_type(2))) float v2f;
typedef __attribute__((ext_vector_type(8))) float v8f;

__global__ __launch_bounds__(THREADS)
void detect_kernel(const float* __restrict__ rois,      // [2048,5]
                   const float* __restrict__ cls_prob,  // [2048,21]
                   const float* __restrict__ bbox_pred, // [2048,84]
                   const float* __restrict__ im_info,   // [3] = H,W,scale
                   const float* __restrict__ gt_boxes,  // [50,5]
                   const int*   __restrict__ num_boxes,
                   float* __restrict__ out)             // [20,2048,5]
{
    __shared__ unsigned long long keyS[N_ROIS];   // 16 KB  sort keys
    __shared__ float4             boxS[N_ROIS];   // 32 KB  decoded boxes
    __shared__ float              scoS[N_ROIS];   //  8 KB  scores
    __shared__ unsigned char      keepS[N_ROIS];  //  2 KB  keep mask
    __shared__ float4             gtS[MAX_GT];    //  1 KB
    __shared__ float              gtAreaS[MAX_GT];
    __shared__ int                nvalidS;

    const int tid = threadIdx.x;
    const int cls = blockIdx.x + 1;               // skip background channel
    const float H = im_info[0], W = im_info[1];
    const float maxx = W - 1.0f, maxy = H - 1.0f;

    // Prefetch this class's strided columns (global_prefetch_b8).
    __builtin_prefetch(cls_prob + cls, 0, 1);
    __builtin_prefetch(bbox_pred + cls * 4, 0, 1);

    if (tid == 0) nvalidS = 0;

    // ---- GT boxes, padded to 64 with degenerate boxes (IoU == 0) ----
    int nb = *num_boxes; nb = nb < 0 ? 0 : (nb > 50 ? 50 : nb);
    if (tid < MAX_GT) {
        float4 g;
        if (tid < nb) {
            g.x = gt_boxes[tid * 5 + 0]; g.y = gt_boxes[tid * 5 + 1];
            g.z = gt_boxes[tid * 5 + 2]; g.w = gt_boxes[tid * 5 + 3];
        } else {
            g.x = g.y = g.z = g.w = -1.0e8f;      // area 0, intersection 0
        }
        gtS[tid] = g;
        gtAreaS[tid] = (g.z - g.x) * (g.w - g.y);
    }

    // ---- Decode + clip + build sort keys ----
    int localValid = 0;
    for (int j = tid; j < N_ROIS; j += THREADS) {
        float x1 = rois[j * 5 + 1], y1 = rois[j * 5 + 2];
        float x2 = rois[j * 5 + 3], y2 = rois[j * 5 + 4];
        float w = x2 - x1 + 1.0f, h = y2 - y1 + 1.0f;
        float cx = x1 + 0.5f * w, cy = y1 + 0.5f * h;
        const float* dp = bbox_pred + j * (4 * N_CLS) + cls * 4;
        float d0 = dp[0] * 0.1f, d1 = dp[1] * 0.1f;
        float d2 = dp[2] * 0.2f, d3 = dp[3] * 0.2f;
        float pcx = d0 * w + cx, pcy = d1 * h + cy;
        float pw = expf(d2) * w, ph = expf(d3) * h;
        float4 b;
        b.x = fminf(fmaxf(pcx - 0.5f * pw, 0.0f), maxx);
        b.y = fminf(fmaxf(pcy - 0.5f * ph, 0.0f), maxy);
        b.z = fminf(fmaxf(pcx + 0.5f * pw, 0.0f), maxx);
        b.w = fminf(fmaxf(pcy + 0.5f * ph, 0.0f), maxy);
        float s = cls_prob[j * N_CLS + cls];
        boxS[j] = b; scoS[j] = s;
        bool valid = s > SCORE_T;
        // key: valid -> (0x7FFFFFFF - bits(s)) descending-score asc order;
        // invalid -> 0xFFFFFFFF (== jnp "+inf" keys). Low 32 bits: stable index.
        unsigned hi = valid ? (0x7FFFFFFFu - __float_as_uint(s)) : 0xFFFFFFFFu;
        keyS[j] = ((unsigned long long)hi << 32) | (unsigned)j;
        localValid += valid ? 1 : 0;
    }
    atomicAdd(&nvalidS, localValid);
    __syncthreads();

    // ---- Bitonic sort (ascending) of 2048 64-bit keys in LDS ----
    for (unsigned k = 2; k <= N_ROIS; k <<= 1) {
        for (unsigned stride = k >> 1; stride > 0; stride >>= 1) {
            for (unsigned i = tid; i < N_ROIS; i += THREADS) {
                unsigned ixj = i ^ stride;
                if (ixj > i) {
                    unsigned long long a = keyS[i], b = keyS[ixj];
                    bool up = ((i & k) == 0);
                    if ((a > b) == up) { keyS[i] = b; keyS[ixj] = a; }
                }
            }
            __syncthreads();
        }
    }

    // ---- Apply permutation in place (gather -> regs -> barrier -> scatter) ----
    float4        pb[PER_T];
    float         ps[PER_T];
    unsigned char pv[PER_T];
#pragma unroll
    for (int q = 0; q < PER_T; ++q) {
        int k2 = tid + q * THREADS;
        unsigned long long key = keyS[k2];
        unsigned idx = (unsigned)(key & 0xFFFFFFFFu);
        pb[q] = boxS[idx];
        ps[q] = scoS[idx];
        pv[q] = ((unsigned)(key >> 32) != 0xFFFFFFFFu) ? 1 : 0;
    }
    __syncthreads();
#pragma unroll
    for (int q = 0; q < PER_T; ++q) {
        int k2 = tid + q * THREADS;
        boxS[k2] = pb[q]; scoS[k2] = ps[q]; keepS[k2] = pv[q];
    }
    __syncthreads();

    const int nv = nvalidS;

    // ---- Greedy NMS: barrier only on active pivots ----
    for (int i = 0; i < nv; ++i) {
        if (keepS[i]) {                    // uniform LDS read
            float4 bi = boxS[i];
            float ai = (bi.z - bi.x) * (bi.w - bi.y);
            for (int j2 = i + 1 + tid; j2 < nv; j2 += THREADS) {
                float4 bj = boxS[j2];
                float aj = (bj.z - bj.x) * (bj.w - bj.y);
                float ltx = fmaxf(bi.x, bj.x), lty = fmaxf(bi.y, bj.y);
                float rbx = fminf(bi.z, bj.z), rby = fminf(bi.w, bj.w);
                float ww = fmaxf(rbx - ltx, 0.0f), hh = fmaxf(rby - lty, 0.0f);
                float inter = ww * hh;
                float iou = inter / (ai + aj - inter + 1e-6f);
                if (iou > NMS_T) keepS[j2] = 0;
            }
            __syncthreads();
        }
    }
    __syncthreads();                       // NMS fully done before dedup writes

    // ---- GT dedup. WMMA computes S[m][n] = areaBox[m] + areaGT[n] exactly:
    //      A(16x4) = [area, 1, 0, 0], B(4x16) = [ones; areaGT; 0; 0].
    {
        const int lane  = tid & 31;
        const int wave  = tid >> 5;
        const int mHalf = (lane >= 16) ? 8 : 0;
        const int l16   = lane & 15;
        for (int tile = wave; tile * 16 < nv; tile += (THREADS / 32)) {
            const int base = tile * 16;
            // A operand (ISA 7.12.2: lanes 0-15 hold K0/K1; lanes 16-31 K2/K3=0)
            float areaM = 0.0f, oneA = 0.0f;
            if (lane < 16) {
                float4 bm = boxS[base + lane];
                areaM = (bm.z - bm.x) * (bm.w - bm.y);
                oneA = 1.0f;
            }
            v2f A; A.x = areaM; A.y = oneA;
            for (int g0 = 0; g0 < MAX_GT; g0 += 16) {
                float4 g = gtS[g0 + l16];
                float gA = gtAreaS[g0 + l16];
                v2f B;
                B.x = (lane < 16) ? 1.0f : 0.0f;   // K0 row: ones
                B.y = (lane < 16) ? gA   : 0.0f;   // K1 row: gt areas
                v8f C = {};
                v8f D = __builtin_amdgcn_wmma_f32_16x16x4_f32(
                    false, A, false, B, (short)0, C, false, false);
#pragma unroll
                for (int r = 0; r < 8; ++r) {
                    int m = base + r + mHalf;      // D vgpr r: M=r (+8 hi half), N=l16
                    float4 bm = boxS[m];
                    float ltx = fmaxf(bm.x, g.x), lty = fmaxf(bm.y, g.y);
                    float rbx = fminf(bm.z, g.z), rby = fminf(bm.w, g.w);
                    float ww = fmaxf(rbx - ltx, 0.0f), hh = fmaxf(rby - lty, 0.0f);
                    float inter = ww * hh;
                    float iou = inter / (D[r] - inter + 1e-6f);
                    if (iou > DEDUP_T) keepS[m] = 0;   // benign same-value race
                }
            }
        }
    }
    __syncthreads();

    // ---- Emit (boxes, score) * mask ----
    const int outBase = blockIdx.x * N_ROIS * 5;
    for (int k2 = tid; k2 < N_ROIS; k2 += THREADS) {
        float4 b = boxS[k2];
        float  s = scoS[k2];
        float  m = keepS[k2] ? 1.0f : 0.0f;
        float* o = out + outBase + k2 * 5;
        o[0] = b.x * m; o[1] = b.y * m; o[2] = b.z * m; o[3] = b.w * m; o[4] = s * m;
    }
}

extern "C" void kernel_launch(void* const* d_in, const int* in_sizes, int n_in,
                              void* d_out, int out_size, void* d_ws, size_t ws_size,
                              hipStream_t stream) {
    (void)in_sizes; (void)n_in; (void)out_size; (void)d_ws; (void)ws_size;
    const float* rois = (const float*)d_in[0];
    const float* cls  = (const float*)d_in[1];
    const float* pred = (const float*)d_in[2];
    const float* imi  = (const float*)d_in[3];
    const float* gt   = (const float*)d_in[4];
    const int*   nbx  = (const int*)d_in[5];
    detect_kernel<<<N_DET, THREADS, 0, stream>>>(rois, cls, pred, imi, gt, nbx,
                                                 (float*)d_out);
}